// ridge_43250320670736
// MI455X (gfx1250) — compile-verified
//
#include <hip/hip_runtime.h>
#include <math.h>

// Problem constants (from the reference): x [N, D] f32, y [N,1] i32, w [D+1,1] f32.
// Outputs concatenated in d_out: pred[N] | loss[1] | K[N*N]   (all f32)
#define NS 8192
#define DF 3072

typedef __attribute__((ext_vector_type(2))) float v2f;
typedef __attribute__((ext_vector_type(8))) float v8f;

// ---------------------------------------------------------------------------
// Pass 1: fused rowsum + ridge matvec via V_WMMA_F32_16X16X4_F32.
//   C(16x16) += A(16x4) * B(4x16), where
//     A = 16-row strip of x over a k-chunk of 4 columns
//     B[:,0] = 1   -> C[:,0] = s[row]   (row sums)
//     B[:,1] = w_k -> C[:,1] = x[row,:].w  (ridge dot)
// One wave per 16-row strip, 4 waves per block. D=3072 -> 768 WMMAs/strip,
// split over two accumulators to break the D->C dependency chain.
// Writes pred = dot + bias, e = exp(s), r = exp(-s).
// ---------------------------------------------------------------------------
__global__ __launch_bounds__(128) void rowsum_pred_kernel(
    const float* __restrict__ x, const float* __restrict__ w,
    float* __restrict__ pred, float* __restrict__ e_out, float* __restrict__ r_out)
{
  const int lane  = threadIdx.x & 31;
  const int wave  = threadIdx.x >> 5;
  const int strip = (blockIdx.x * 4 + wave) * 16;   // first row of this wave's strip

  const int n = lane & 15;   // B/C column index for this lane
  const int g = lane >> 4;   // lane group: A/B hold K = {0,1} (g=0) or {2,3} (g=1)

  // Branch-free B construction masks: column 0 -> ones, column 1 -> w, else 0.
  const float sel0 = (n == 0) ? 1.0f : 0.0f;
  const float sel1 = (n == 1) ? 1.0f : 0.0f;

  // A-matrix per ISA layout: lane (g,n) supplies x[strip+n][k + 2g + {0,1}]
  const float* __restrict__ xrow = x + (size_t)(strip + n) * DF + 2 * g;
  const float* __restrict__ wk   = w + 2 * g;

  v8f acc0 = {0.f, 0.f, 0.f, 0.f, 0.f, 0.f, 0.f, 0.f};
  v8f acc1 = {0.f, 0.f, 0.f, 0.f, 0.f, 0.f, 0.f, 0.f};

  for (int k = 0; k < DF; k += 8) {
    // chunk k .. k+3
    v2f a0 = *(const v2f*)(xrow + k);         // 8B aligned (even float offsets)
    v2f wv0 = *(const v2f*)(wk + k);          // 12 KB array -> L0 resident
    v2f b0;
    b0.x = sel0 + sel1 * wv0.x;
    b0.y = sel0 + sel1 * wv0.y;
    acc0 = __builtin_amdgcn_wmma_f32_16x16x4_f32(
        false, a0, false, b0, (short)0, acc0, false, false);

    // chunk k+4 .. k+7 (independent accumulator)
    v2f a1 = *(const v2f*)(xrow + k + 4);
    v2f wv1 = *(const v2f*)(wk + k + 4);
    v2f b1;
    b1.x = sel0 + sel1 * wv1.x;
    b1.y = sel0 + sel1 * wv1.y;
    acc1 = __builtin_amdgcn_wmma_f32_16x16x4_f32(
        false, a1, false, b1, (short)0, acc1, false, false);
  }

  v8f c = acc0 + acc1;
  const float bias = w[DF];   // weight[D] is the bias (ones column of xb)

  // C layout: lanes 0-15 hold M = v (VGPR v), lanes 16-31 hold M = v + 8.
  if (n == 0) {               // column 0 = row sums s
#pragma unroll
    for (int v = 0; v < 8; ++v) {
      const int row = strip + g * 8 + v;
      const float s = c[v];
      e_out[row] = expf(s);    // e[j] = exp(s[j])
      r_out[row] = expf(-s);   // r[i] = exp(-s[i])
    }
  } else if (n == 1) {        // column 1 = x.w
#pragma unroll
    for (int v = 0; v < 8; ++v) {
      const int row = strip + g * 8 + v;
      pred[row] = c[v] + bias;
    }
  }
}

// ---------------------------------------------------------------------------
// Pass 2: loss = 0.5*sqrt(sum((y-pred)^2))/N + sum(|w|). One block, LDS tree.
// ---------------------------------------------------------------------------
__global__ __launch_bounds__(256) void loss_kernel(
    const float* __restrict__ pred, const int* __restrict__ y,
    const float* __restrict__ w, float* __restrict__ out_loss)
{
  __shared__ float red[256];
  const int tid = threadIdx.x;

  float sq = 0.f;
  for (int i = tid; i < NS; i += 256) {
    const float d = (float)y[i] - pred[i];
    sq += d * d;
  }
  float ab = 0.f;
  for (int d = tid; d <= DF; d += 256) ab += fabsf(w[d]);

  red[tid] = sq;
  __syncthreads();
  for (int s = 128; s > 0; s >>= 1) {
    if (tid < s) red[tid] += red[tid + s];
    __syncthreads();
  }
  const float total_sq = red[0];
  __syncthreads();

  red[tid] = ab;
  __syncthreads();
  for (int s = 128; s > 0; s >>= 1) {
    if (tid < s) red[tid] += red[tid + s];
    __syncthreads();
  }

  if (tid == 0)
    out_loss[0] = 0.5f * sqrtf(total_sq) / (float)NS + red[0];
}

// ---------------------------------------------------------------------------
// Pass 3: K[i,j] = r[i] * e[j]  (rank-1 outer product, 256 MB of stores).
// One block per row; e[] (32 KB) hits L0/L2. Non-temporal b32 stores: K is
// write-once and larger than L2, and the K base is only 4B-aligned
// (d_out + 8193 floats), so coalesced b32 NT stores (wave32 x 4B = one full
// 128B line per instruction) are the bandwidth-correct choice.
// ---------------------------------------------------------------------------
__global__ __launch_bounds__(256) void kmat_kernel(
    const float* __restrict__ e, const float* __restrict__ r,
    float* __restrict__ K)
{
  const int i = blockIdx.x;
  const float ri = r[i];
  float* __restrict__ row = K + (size_t)i * NS;
  for (int j = threadIdx.x; j < NS; j += 256) {
    __builtin_nontemporal_store(ri * e[j], row + j);
  }
}

// ---------------------------------------------------------------------------
extern "C" void kernel_launch(void* const* d_in, const int* in_sizes, int n_in,
                              void* d_out, int out_size, void* d_ws, size_t ws_size,
                              hipStream_t stream) {
  (void)in_sizes; (void)n_in; (void)out_size; (void)ws_size;

  const float* x = (const float*)d_in[0];   // [8192, 3072] f32
  const int*   y = (const int*)d_in[1];     // [8192, 1]    i32
  const float* w = (const float*)d_in[2];   // [3073, 1]    f32

  float* out   = (float*)d_out;
  float* pred  = out;              // [8192]
  float* lossp = out + NS;         // [1]
  float* K     = out + NS + 1;     // [8192*8192]

  float* e = (float*)d_ws;         // [8192] exp(s)
  float* r = e + NS;               // [8192] exp(-s)

  rowsum_pred_kernel<<<NS / 64, 128, 0, stream>>>(x, w, pred, e, r);
  loss_kernel<<<1, 256, 0, stream>>>(pred, y, w, lossp);
  kmat_kernel<<<NS, 256, 0, stream>>>(e, r, K);
}